// PtModule_76166950027565
// MI455X (gfx1250) — compile-verified
//
#include <hip/hip_runtime.h>

// y = min(((x + 1) * 0.75)^2, 10)  -- pure streaming elementwise, fp32 in/out.
// HBM-bound: 512 MiB traffic, ~22us floor at 23.3 TB/s.
// Vec kernel: NO bounds checks (full tiles only) -> 4 clause'd b128 NT loads,
// one wait, 48 VALU, 4 b128 NT stores. Tail handled by separate kernel.

typedef float float4v __attribute__((ext_vector_type(4)));

#define BLOCK 256
#define UNROLL 4
#define TILE_F4 (BLOCK * UNROLL)      // float4's per block-tile = 1024
#define TILE_F  (TILE_F4 * 4)         // floats per block-tile  = 4096

__device__ __forceinline__ float mapf(float x) {
    float t = __builtin_fmaf(x, 0.75f, 0.75f);  // (x+1)*0.75
    float y = t * t;
    return __builtin_fminf(y, 10.0f);
}

// Full-tile kernel: every access in range by construction. No EXEC branching,
// so all 4 global_load_b128 issue back-to-back (single s_clause) before the
// first s_wait_loadcnt — max memory-level parallelism per wave.
__global__ __launch_bounds__(BLOCK) void elemwise_vec_kernel(
    const float* __restrict__ x, float* __restrict__ out) {
    const float4v* __restrict__ xv =
        (const float4v*)x + (long long)blockIdx.x * TILE_F4 + threadIdx.x;
    float4v* __restrict__ ov =
        (float4v*)out + (long long)blockIdx.x * TILE_F4 + threadIdx.x;

    float4v v[UNROLL];
#pragma unroll
    for (int u = 0; u < UNROLL; ++u) {
        // constant byte offset u*BLOCK*16 folds into the IOFFSET field
        v[u] = __builtin_nontemporal_load(&xv[u * BLOCK]);
    }

#pragma unroll
    for (int u = 0; u < UNROLL; ++u) {
        float4v r;
#pragma unroll
        for (int k = 0; k < 4; ++k) {
            r[k] = mapf(v[u][k]);
        }
        __builtin_nontemporal_store(r, &ov[u * BLOCK]);
    }
}

// Guarded scalar tail for elements past the last full tile (not launched for
// 8192^2, which divides evenly into 16384 tiles).
__global__ __launch_bounds__(BLOCK) void elemwise_tail_kernel(
    const float* __restrict__ x, float* __restrict__ out,
    long long start, long long n) {
    long long i = start + (long long)blockIdx.x * BLOCK + threadIdx.x;
    if (i < n) {
        out[i] = mapf(x[i]);
    }
}

extern "C" void kernel_launch(void* const* d_in, const int* in_sizes, int n_in,
                              void* d_out, int out_size, void* d_ws, size_t ws_size,
                              hipStream_t stream) {
    (void)n_in; (void)d_ws; (void)ws_size; (void)out_size;

    const float* x = (const float*)d_in[0];
    float* out = (float*)d_out;

    long long n = (long long)in_sizes[0];   // 8192*8192 = 67,108,864
    long long full_tiles = n / TILE_F;      // 16384 for this shape
    long long done = full_tiles * TILE_F;
    long long rem = n - done;

    if (full_tiles > 0) {
        elemwise_vec_kernel<<<(unsigned)full_tiles, BLOCK, 0, stream>>>(x, out);
    }
    if (rem > 0) {
        unsigned tail_blocks = (unsigned)((rem + BLOCK - 1) / BLOCK);
        elemwise_tail_kernel<<<tail_blocks, BLOCK, 0, stream>>>(x, out, done, n);
    }
}